// FixedGATLayer_56788057588237
// MI455X (gfx1250) — compile-verified
//
#include <hip/hip_runtime.h>
#include <math.h>

// ---------------------------------------------------------------------------
// Fixed GAT layer for MI455X (gfx1250, wave32).
//   inputs:  h[N,128] f32, W_w[64,128] f32, W_b[64] f32, a_w[1,128] f32,
//            a_b[1] f32, edge_index[2,E] int
//   output:  leaky_relu(scatter_add(attn * Wh[src] -> tgt))  [N,64] f32
// ---------------------------------------------------------------------------

typedef float  v2f __attribute__((ext_vector_type(2)));
typedef float  v8f __attribute__((ext_vector_type(8)));

#define IN_DIM   128
#define OUT_DIM  64
#define SLOPE    0.2f

// LDS B layout: row r = k>>1 holds the (k, k+1) pair interleaved per column:
//   ldsB2[r*B_STRIDE + n*2 + (k&1)]
// B_STRIDE = 160 dwords (pad 32) so the two wave halves (kk=k vs kk=k+2) hit
// disjoint bank halves: lanes 0-15 -> banks 0..31, lanes 16-31 -> banks 32..63.
#define B_STRIDE 160

__device__ __forceinline__ float lrelu(float x) {
    return x > 0.0f ? x : SLOPE * x;
}

// ---------------------------------------------------------------------------
// K1: Wh = h @ W^T + b via V_WMMA_F32_16X16X4_F32, fused with
//     s_src = Wh @ a[:64], s_tgt = Wh @ a[64:]  (per-node attention halves).
// One wave computes a 16(M) x 64(N) tile: 4 accumulators, 32 K-steps of 4.
// W^T is staged into LDS once per 8-wave workgroup in pair-interleaved form
// so every B operand is a single aligned ds_load_b64 into an even VGPR pair.
// ---------------------------------------------------------------------------
__global__ __launch_bounds__(256)
void gat_gemm_wmma(const float* __restrict__ h,
                   const float* __restrict__ Ww,   // [64,128] row-major (out,in)
                   const float* __restrict__ Wb,   // [64]
                   const float* __restrict__ aw,   // [128]
                   float* __restrict__ Wh,         // [N,64]
                   float* __restrict__ s_src,      // [N]
                   float* __restrict__ s_tgt,      // [N]
                   int nNodes)
{
    __shared__ float ldsB2[(IN_DIM / 2) * B_STRIDE];   // 40 KB

    const int tid = threadIdx.x;
    // stage W into LDS, pair-interleaved: ldsB2[(k>>1)*B_STRIDE + n*2 + (k&1)]
    // reads are coalesced along k within each row n of Ww.
    for (int idx = tid; idx < IN_DIM * OUT_DIM; idx += 256) {
        int n = idx >> 7;          // 0..63
        int k = idx & 127;         // 0..127
        ldsB2[(k >> 1) * B_STRIDE + n * 2 + (k & 1)] = Ww[idx];
    }
    __syncthreads();

    const int wave   = tid >> 5;
    const int lane   = tid & 31;
    const int lane16 = lane & 15;
    const bool hiHalf = (lane & 16) != 0;

    const int mTile = blockIdx.x * 8 + wave;   // 16-row tile index
    const int mTiles = (nNodes + 15) >> 4;
    if (mTile >= mTiles) return;
    const int m0 = mTile * 16;

    // per-lane column constants for the 4 N-subtiles
    float biasv[4], aSv[4], aTv[4];
#pragma unroll
    for (int nt = 0; nt < 4; ++nt) {
        int col = nt * 16 + lane16;
        biasv[nt] = Wb[col];
        aSv[nt]   = aw[col];
        aTv[nt]   = aw[OUT_DIM + col];
    }

    v8f acc[4] = {};
    // A layout (ISA 7.12.2, 32-bit A 16x4): lanes 0-15 -> K=0,1 ; lanes 16-31 -> K=2,3
    const int mRow = m0 + lane16;
    const int kOff = hiHalf ? 2 : 0;
    // LDS base for this lane: row (kk>>1), column pair n*2 (kk even -> 8B aligned)
    const float* ldsBase = ldsB2 + (kOff >> 1) * B_STRIDE + lane16 * 2;

#pragma unroll 4
    for (int k = 0; k < IN_DIM; k += 4) {
        // aligned 8-byte load of (K, K+1) pair for this lane's row of A
        v2f a = *(const v2f*)(h + (size_t)mRow * IN_DIM + (k + kOff));
        const float* bp = ldsBase + (k >> 1) * B_STRIDE;
#pragma unroll
        for (int nt = 0; nt < 4; ++nt) {
            v2f b = *(const v2f*)(bp + nt * 32);   // ds_load_b64, conflict-free
            acc[nt] = __builtin_amdgcn_wmma_f32_16x16x4_f32(
                false, a, false, b, (short)0, acc[nt], false, false);
        }
    }

    // Epilogue: bias add, store Wh, and fused s_src/s_tgt row reductions.
    // D layout: VGPR i -> (row m0+i, col=lane) lanes 0-15 ; (row m0+8+i) lanes 16-31
#pragma unroll
    for (int i = 0; i < 8; ++i) {
        int row = m0 + i + (hiHalf ? 8 : 0);
        float pS = 0.0f, pT = 0.0f;
#pragma unroll
        for (int nt = 0; nt < 4; ++nt) {
            float val = acc[nt][i] + biasv[nt];
            Wh[(size_t)row * OUT_DIM + nt * 16 + lane16] = val;
            pS += val * aSv[nt];
            pT += val * aTv[nt];
        }
        // reduce across the 16 lanes of each half (xor masks stay within halves)
#pragma unroll
        for (int m = 8; m >= 1; m >>= 1) {
            pS += __shfl_xor(pS, m, 32);
            pT += __shfl_xor(pT, m, 32);
        }
        if (lane16 == 0) {
            s_src[row] = pS;
            s_tgt[row] = pT;
        }
    }
}

// ---------------------------------------------------------------------------
// K2: edge logits + per-block max partials
// ---------------------------------------------------------------------------
__global__ __launch_bounds__(256)
void gat_edge_logits(const float* __restrict__ s_src,
                     const float* __restrict__ s_tgt,
                     const int* __restrict__ srcIdx,
                     const int* __restrict__ tgtIdx,
                     const float* __restrict__ ab,
                     float* __restrict__ logits,
                     float* __restrict__ blockMax,
                     int nEdges, int nNodes)
{
    __shared__ float red[256];
    int e = blockIdx.x * 256 + threadIdx.x;
    float l = -3.402823e38f;
    if (e < nEdges) {
        int s = srcIdx[e]; int t = tgtIdx[e];
        s = (s >= 0 && s < nNodes) ? s : 0;
        t = (t >= 0 && t < nNodes) ? t : 0;
        l = lrelu(s_src[s] + s_tgt[t] + ab[0]);
        logits[e] = l;
    }
    red[threadIdx.x] = l;
    __syncthreads();
    for (int w = 128; w >= 1; w >>= 1) {
        if (threadIdx.x < w) red[threadIdx.x] = fmaxf(red[threadIdx.x], red[threadIdx.x + w]);
        __syncthreads();
    }
    if (threadIdx.x == 0) blockMax[blockIdx.x] = red[0];
}

__global__ __launch_bounds__(256)
void gat_reduce_max(const float* __restrict__ partials, int n, float* __restrict__ gmax)
{
    __shared__ float red[256];
    float m = -3.402823e38f;
    for (int i = threadIdx.x; i < n; i += 256) m = fmaxf(m, partials[i]);
    red[threadIdx.x] = m;
    __syncthreads();
    for (int w = 128; w >= 1; w >>= 1) {
        if (threadIdx.x < w) red[threadIdx.x] = fmaxf(red[threadIdx.x], red[threadIdx.x + w]);
        __syncthreads();
    }
    if (threadIdx.x == 0) *gmax = red[0];
}

// ---------------------------------------------------------------------------
// K3: w = exp(logit - gmax), per-block sum partials (overwrites logits)
// ---------------------------------------------------------------------------
__global__ __launch_bounds__(256)
void gat_edge_exp(float* __restrict__ logits,
                  const float* __restrict__ gmax,
                  float* __restrict__ blockSum,
                  int nEdges)
{
    __shared__ float red[256];
    int e = blockIdx.x * 256 + threadIdx.x;
    float w = 0.0f;
    if (e < nEdges) {
        w = __expf(logits[e] - *gmax);
        logits[e] = w;
    }
    red[threadIdx.x] = w;
    __syncthreads();
    for (int s = 128; s >= 1; s >>= 1) {
        if (threadIdx.x < s) red[threadIdx.x] += red[threadIdx.x + s];
        __syncthreads();
    }
    if (threadIdx.x == 0) blockSum[blockIdx.x] = red[0];
}

__global__ __launch_bounds__(256)
void gat_reduce_sum(const float* __restrict__ partials, int n, float* __restrict__ gsum)
{
    __shared__ float red[256];
    float s = 0.0f;
    for (int i = threadIdx.x; i < n; i += 256) s += partials[i];
    red[threadIdx.x] = s;
    __syncthreads();
    for (int w = 128; w >= 1; w >>= 1) {
        if (threadIdx.x < w) red[threadIdx.x] += red[threadIdx.x + w];
        __syncthreads();
    }
    if (threadIdx.x == 0) *gsum = red[0];
}

// ---------------------------------------------------------------------------
// K4: zero the accumulation buffer (d_out), deterministic per call
// ---------------------------------------------------------------------------
__global__ __launch_bounds__(256)
void gat_zero(float* __restrict__ p, int n)
{
    int i = blockIdx.x * 256 + threadIdx.x;
    if (i < n) p[i] = 0.0f;
}

// ---------------------------------------------------------------------------
// K5: scatter-add  messages[tgt] += (w/gsum) * Wh[src]
// 64 threads per edge -> contiguous loads of Wh[src,:] and contiguous
// global_atomic_add_f32 to messages[tgt,:].
// ---------------------------------------------------------------------------
__global__ __launch_bounds__(256)
void gat_scatter(const float* __restrict__ Wh,
                 const float* __restrict__ w,
                 const float* __restrict__ gsum,
                 const int* __restrict__ srcIdx,
                 const int* __restrict__ tgtIdx,
                 float* __restrict__ msg,
                 int nEdges, int nNodes)
{
    unsigned gid = blockIdx.x * 256u + threadIdx.x;
    unsigned e = gid >> 6;
    unsigned d = gid & 63u;
    if (e >= (unsigned)nEdges) return;
    int s = srcIdx[e]; int t = tgtIdx[e];
    s = (s >= 0 && s < nNodes) ? s : 0;
    t = (t >= 0 && t < nNodes) ? t : 0;
    float attn = w[e] * __frcp_rn(*gsum);
    float val  = attn * Wh[(size_t)s * OUT_DIM + d];
#if defined(__HIP_DEVICE_COMPILE__)
    unsafeAtomicAdd(&msg[(size_t)t * OUT_DIM + d], val);   // global_atomic_add_f32
#else
    atomicAdd(&msg[(size_t)t * OUT_DIM + d], val);
#endif
}

// ---------------------------------------------------------------------------
// K6: in-place leaky ReLU on the accumulated messages
// ---------------------------------------------------------------------------
__global__ __launch_bounds__(256)
void gat_lrelu(float* __restrict__ p, int n)
{
    int i = blockIdx.x * 256 + threadIdx.x;
    if (i < n) p[i] = lrelu(p[i]);
}

// ---------------------------------------------------------------------------
extern "C" void kernel_launch(void* const* d_in, const int* in_sizes, int n_in,
                              void* d_out, int out_size, void* d_ws, size_t ws_size,
                              hipStream_t stream)
{
    const float* h  = (const float*)d_in[0];
    const float* Ww = (const float*)d_in[1];
    const float* Wb = (const float*)d_in[2];
    const float* aw = (const float*)d_in[3];
    const float* ab = (const float*)d_in[4];
    const int*   ei = (const int*)d_in[5];

    const int nNodes = in_sizes[0] / IN_DIM;       // 100000
    const int nEdges = in_sizes[5] / 2;            // 1000000
    const int* srcIdx = ei;
    const int* tgtIdx = ei + nEdges;

    // workspace layout (floats)
    float* ws     = (float*)d_ws;
    float* Wh     = ws;                                  // nNodes*64
    float* s_src  = Wh    + (size_t)nNodes * OUT_DIM;    // nNodes
    float* s_tgt  = s_src + nNodes;                      // nNodes
    float* logits = s_tgt + nNodes;                      // nEdges
    float* parts  = logits + nEdges;                     // up to 4096
    float* gmax   = parts + 4096;
    float* gsum   = gmax + 1;

    float* msg = (float*)d_out;                          // accumulate in d_out

    const int mTiles     = (nNodes + 15) / 16;
    const int gemmBlocks = (mTiles + 7) / 8;
    const int edgeBlocks = (nEdges + 255) / 256;
    const int outElems   = nNodes * OUT_DIM;
    const int outBlocks  = (outElems + 255) / 256;
    const int scatBlocks = (int)(((size_t)nEdges * 64 + 255) / 256);

    gat_gemm_wmma<<<gemmBlocks, 256, 0, stream>>>(h, Ww, Wb, aw, Wh, s_src, s_tgt, nNodes);
    gat_edge_logits<<<edgeBlocks, 256, 0, stream>>>(s_src, s_tgt, srcIdx, tgtIdx, ab,
                                                    logits, parts, nEdges, nNodes);
    gat_reduce_max<<<1, 256, 0, stream>>>(parts, edgeBlocks, gmax);
    gat_edge_exp<<<edgeBlocks, 256, 0, stream>>>(logits, gmax, parts, nEdges);
    gat_reduce_sum<<<1, 256, 0, stream>>>(parts, edgeBlocks, gsum);
    gat_zero<<<outBlocks, 256, 0, stream>>>(msg, outElems);
    gat_scatter<<<scatBlocks, 256, 0, stream>>>(Wh, logits, gsum, srcIdx, tgtIdx,
                                                msg, nEdges, nNodes);
    gat_lrelu<<<outBlocks, 256, 0, stream>>>(msg, outElems);
}